// MambaLMBlock_37134287241894
// MI455X (gfx1250) — compile-verified
//
#include <hip/hip_runtime.h>
#include <hip/hip_bf16.h>
#include <math.h>

typedef __attribute__((ext_vector_type(2))) float v2f;
typedef __attribute__((ext_vector_type(8))) float v8f;
typedef __attribute__((ext_vector_type(4))) unsigned int v4u;
typedef __attribute__((ext_vector_type(8))) int v8i;
typedef __attribute__((ext_vector_type(4))) int v4i;

#define D_MODEL 1024
#define D_INNER 2048
#define D_STATE 16
#define D_CONV  4
#define DT_RANK 64
#define B_SZ    2
#define SEQ     2048
#define NROW    (B_SZ * SEQ)     // 4096
#define EPS_F   1e-5f
#define DBL_N   (DT_RANK + 2 * D_STATE)  // 96

#define LDS_STRIDE 33            // 32 dwords + 1 pad dword (TDM pad feature)

// ---------------------------------------------------------------------------
// Tensor Data Mover: issue a 2D tile load (tile_rows x 32 f32) into LDS.
// D# per CDNA5 ISA 8.3/8.4:
//   group0: [1:0]count=1 | lds_addr | global_addr[56:0] | type=2
//   group1: data_size=4B, pad_enable, pad_interval=32dw, pad_amount=1dw,
//           tensor_dim0/1 (OOB zero-fill), tile_dim0=32, tile_dim1=rows,
//           tensor_dim0_stride = row stride (elems)
// groups 2/3 (and trailing group) zero: <=2D tensor.
// ---------------------------------------------------------------------------
__device__ __forceinline__ void tdm_load_tile32(
    unsigned lds_off, const void* gaddr, unsigned row_len_avail,
    unsigned rows_avail, unsigned tile_rows, unsigned row_stride_elems) {
  unsigned long long ga = (unsigned long long)(uintptr_t)gaddr;
  v4u g0;
  g0.x = 1u;                                        // count=1, user descriptor
  g0.y = lds_off;                                   // lds_addr (bytes)
  g0.z = (unsigned)(ga & 0xffffffffu);              // global_addr[31:0]
  g0.w = (unsigned)((ga >> 32) & 0x1ffffffu) | (2u << 30);  // addr[56:32]|type=2

  v8i g1;
  unsigned w0 = (2u << 16)      // data_size = 4 bytes
              | (1u << 20)      // pad_enable
              | (4u << 22)      // pad_interval: 2^(4+1)=32 DWORDs
              | (0u << 25);     // pad_amount: 1 DWORD
  g1[0] = (int)w0;
  g1[1] = (int)((row_len_avail & 0xffffu) << 16);   // tensor_dim0[15:0]
  g1[2] = (int)(((row_len_avail >> 16) & 0xffffu) |
                ((rows_avail & 0xffffu) << 16));    // dim0[31:16] | dim1[15:0]
  g1[3] = (int)(((rows_avail >> 16) & 0xffffu) |
                (32u << 16));                       // dim1[31:16] | tile_dim0=32
  g1[4] = (int)(tile_rows & 0xffffu);               // tile_dim1 (tile_dim2=0)
  g1[5] = (int)row_stride_elems;                    // tensor_dim0_stride[31:0]
  g1[6] = 0;                                        // stride hi / dim1_stride lo
  g1[7] = 0;

  v4i z4 = {0, 0, 0, 0};
  v8i z8 = {0, 0, 0, 0, 0, 0, 0, 0};
  __builtin_amdgcn_tensor_load_to_lds(g0, g1, z4, z4, z8, 0);
}

__device__ __forceinline__ unsigned lds_off_of(const void* p) {
  return (unsigned)(unsigned long long)(uintptr_t)p;  // flat LDS addr low 32 = LDS byte addr
}

// ---------------------------------------------------------------------------
// RMSNorm: one block per row of (NROW x D_MODEL)
// ---------------------------------------------------------------------------
__global__ __launch_bounds__(256) void rmsnorm_kernel(
    const float* __restrict__ x, const float* __restrict__ w,
    float* __restrict__ out) {
  int row = blockIdx.x;
  const float* xr = x + (size_t)row * D_MODEL;
  float* orow = out + (size_t)row * D_MODEL;

  float s = 0.f;
  for (int i = threadIdx.x; i < D_MODEL; i += 256) {
    float v = xr[i];
    s += v * v;
  }
  for (int o = 16; o > 0; o >>= 1) s += __shfl_xor(s, o, 32);
  __shared__ float partial[8];
  __shared__ float rstd_s;
  int wid = threadIdx.x >> 5, lane = threadIdx.x & 31;
  if (lane == 0) partial[wid] = s;
  __syncthreads();
  if (threadIdx.x == 0) {
    float t = 0.f;
    for (int i = 0; i < 8; i++) t += partial[i];
    rstd_s = rsqrtf(t / (float)D_MODEL + EPS_F);
  }
  __syncthreads();
  float r = rstd_s;
  for (int i = threadIdx.x; i < D_MODEL; i += 256) {
    orow[i] = xr[i] * r * w[i];
  }
}

// ---------------------------------------------------------------------------
// WMMA fp32 GEMM with TDM staging:  C[M,N] = A[M,K(lda)] @ W[N,K]^T
//   EPI 0: plain store
//   EPI 1: softplus(acc + aux[col])      (dt_proj)
//   EPI 2: acc + aux[row*N + col]        (out_proj residual)
// Block: 256 threads = 8 waves; block tile 64(M) x 64(N); K step 32.
// Wave (wm = wave&3, wn = wave>>2) computes 16x32: two 16x16 WMMA tiles
// sharing one A fragment.  LDS tiles staged by the Tensor Data Mover with
// double buffering; TDM pads rows to 33 dwords (bank-conflict-free) and
// zero-fills out-of-range rows (handles N=96 edge).
// Fragment layouts per CDNA5 ISA 7.12.2 (f32, K=4):
//   A 16x4 : lanes 0..15 = M, vgpr v holds K = v + 2*(lane>=16)
//   B 4x16 : lanes 0..15 = N, vgpr v holds K = v + 2*(lane>=16)
//   C 16x16: vgpr r -> M = r + 8*(lane>=16), N = lane&15
// ---------------------------------------------------------------------------
template <int EPI>
__global__ __launch_bounds__(256) void wmma_gemm_f32(
    const float* __restrict__ A, const float* __restrict__ W,
    float* __restrict__ C, int M, int N, int K, int lda,
    const float* __restrict__ aux) {
  __shared__ float As[2][64 * LDS_STRIDE];   // [buf][row*33 + k]
  __shared__ float Bs[2][64 * LDS_STRIDE];   // [buf][n*33 + k]

  const int m0 = blockIdx.y * 64;
  const int n0 = blockIdx.x * 64;
  const int tid = threadIdx.x;
  const int wave = tid >> 5;
  const int lane = tid & 31;
  const int wm = wave & 3;        // 4 tiles along M
  const int wn = wave >> 2;       // 2 half-tiles (32 cols) along N
  const int frow = lane & 15;
  const int hi = lane >> 4;

  v8f acc0 = {};
  v8f acc1 = {};

  // Prime buffer 0 (k0 = 0).
  if (wave == 0) {
    tdm_load_tile32(lds_off_of(&As[0][0]), A + (size_t)m0 * lda,
                    (unsigned)lda, (unsigned)(M - m0), 64u, (unsigned)lda);
    tdm_load_tile32(lds_off_of(&Bs[0][0]), W + (size_t)n0 * K,
                    (unsigned)K, (unsigned)(N - n0), 64u, (unsigned)K);
  }

  for (int k0 = 0; k0 < K; k0 += 32) {
    const int buf = (k0 >> 5) & 1;
    if (wave == 0) {
      if (k0 + 32 < K) {
        // Kick off next tile DMA into the other buffer, then wait only for
        // the current buffer's two loads (in-order completion).
        tdm_load_tile32(lds_off_of(&As[buf ^ 1][0]),
                        A + (size_t)m0 * lda + (k0 + 32), (unsigned)(lda - k0 - 32),
                        (unsigned)(M - m0), 64u, (unsigned)lda);
        tdm_load_tile32(lds_off_of(&Bs[buf ^ 1][0]),
                        W + (size_t)n0 * K + (k0 + 32), (unsigned)(K - k0 - 32),
                        (unsigned)(N - n0), 64u, (unsigned)K);
        __builtin_amdgcn_s_wait_tensorcnt((short)2);
      } else {
        __builtin_amdgcn_s_wait_tensorcnt((short)0);
      }
    }
    __syncthreads();

    const float* ar = &As[buf][(wm * 16 + frow) * LDS_STRIDE + 2 * hi];
    const float* br0 = &Bs[buf][(wn * 32 + frow) * LDS_STRIDE + 2 * hi];
    const float* br1 = br0 + 16 * LDS_STRIDE;
#pragma unroll
    for (int kk = 0; kk < 32; kk += 4) {
      v2f a, b0, b1;
      a.x = ar[kk];
      a.y = ar[kk + 1];
      b0.x = br0[kk];
      b0.y = br0[kk + 1];
      b1.x = br1[kk];
      b1.y = br1[kk + 1];
      acc0 = __builtin_amdgcn_wmma_f32_16x16x4_f32(false, a, false, b0,
                                                   (short)0, acc0, false, false);
      acc1 = __builtin_amdgcn_wmma_f32_16x16x4_f32(false, a, false, b1,
                                                   (short)0, acc1, false, false);
    }
    __syncthreads();
  }

  // --- epilogue ---
  const int col0 = n0 + wn * 32 + frow;
  const int col1 = col0 + 16;
#pragma unroll
  for (int r = 0; r < 8; r++) {
    int orow = m0 + wm * 16 + r + 8 * hi;
    float v0 = acc0[r];
    float v1 = acc1[r];
    if (EPI == 1) {
      v0 += aux[col0 < N ? col0 : 0];
      v1 += aux[col1 < N ? col1 : 0];
      v0 = (v0 > 20.f) ? v0 : log1pf(expf(v0));
      v1 = (v1 > 20.f) ? v1 : log1pf(expf(v1));
    } else if (EPI == 2) {
      if (col0 < N) v0 += aux[(size_t)orow * N + col0];
      if (col1 < N) v1 += aux[(size_t)orow * N + col1];
    }
    if (col0 < N) C[(size_t)orow * N + col0] = v0;
    if (col1 < N) C[(size_t)orow * N + col1] = v1;
  }
}

// ---------------------------------------------------------------------------
// Depthwise causal conv(4) + bias + SiLU.
// ---------------------------------------------------------------------------
__global__ __launch_bounds__(256) void conv_silu_kernel(
    const float* __restrict__ xz, const float* __restrict__ cw,
    const float* __restrict__ cb, float* __restrict__ xact) {
  int idx = blockIdx.x * 256 + threadIdx.x;
  if (idx >= NROW * D_INNER) return;
  int d = idx & (D_INNER - 1);
  int rowg = idx >> 11;
  int t = rowg & (SEQ - 1);
  int b = rowg >> 11;

  float acc = cb[d];
#pragma unroll
  for (int j = 0; j < D_CONV; j++) {
    int tt = t - (D_CONV - 1) + j;
    if (tt >= 0) {
      float xv = xz[(size_t)(b * SEQ + tt) * (2 * D_INNER) + d];
      acc += cw[d * D_CONV + j] * xv;
    }
  }
  xact[idx] = acc / (1.f + expf(-acc));  // silu
}

// ---------------------------------------------------------------------------
// Selective scan.  One wave per (batch, channel-pair): lane = n + 16*sub,
// d = 2*dpair + sub.  h' = h*exp(dt*A) + (dt*x)*B ; y = sum_n h*C
// ---------------------------------------------------------------------------
__global__ __launch_bounds__(256) void scan_kernel(
    const float* __restrict__ xact, const float* __restrict__ dtb,
    const float* __restrict__ dbl, const float* __restrict__ A_log,
    float* __restrict__ ys) {
  int wave_global = blockIdx.x * 8 + (threadIdx.x >> 5);  // 2048 total
  int lane = threadIdx.x & 31;
  int n = lane & 15;
  int sub = lane >> 4;
  int b = wave_global >> 10;
  int dpair = wave_global & 1023;
  int d = dpair * 2 + sub;

  const float Acoef = -expf(A_log[d * D_STATE + n]);
  float h = 0.f;

  for (int t = 0; t < SEQ; t++) {
    size_t rowg = (size_t)(b * SEQ + t);
    float dtv = dtb[rowg * D_INNER + d];
    float xv = xact[rowg * D_INNER + d];
    float Bv = dbl[rowg * DBL_N + DT_RANK + n];
    float Cv = dbl[rowg * DBL_N + DT_RANK + D_STATE + n];
    if (t + 1 < SEQ) {  // hide next-step latency
      __builtin_prefetch(&dtb[(rowg + 1) * D_INNER + d], 0, 0);
      __builtin_prefetch(&dbl[(rowg + 1) * DBL_N + DT_RANK + n], 0, 0);
    }
    float dA = expf(dtv * Acoef);
    h = h * dA + (dtv * xv) * Bv;
    float y = h * Cv;
    y += __shfl_xor(y, 1, 16);
    y += __shfl_xor(y, 2, 16);
    y += __shfl_xor(y, 4, 16);
    y += __shfl_xor(y, 8, 16);
    if (n == 0) ys[rowg * D_INNER + d] = y;
  }
}

// ---------------------------------------------------------------------------
// Gating: g = (ys + xact*D) * silu(z).  In-place on ys.
// ---------------------------------------------------------------------------
__global__ __launch_bounds__(256) void gate_kernel(
    const float* __restrict__ ys, const float* __restrict__ xact,
    const float* __restrict__ xz, const float* __restrict__ Dw,
    float* __restrict__ g) {
  int idx = blockIdx.x * 256 + threadIdx.x;
  if (idx >= NROW * D_INNER) return;
  int d = idx & (D_INNER - 1);
  size_t rowg = (size_t)(idx >> 11);
  float z = xz[rowg * (2 * D_INNER) + D_INNER + d];
  float y = ys[idx] + xact[idx] * Dw[d];
  g[idx] = y * (z / (1.f + expf(-z)));
}

// ---------------------------------------------------------------------------
extern "C" void kernel_launch(void* const* d_in, const int* in_sizes, int n_in,
                              void* d_out, int out_size, void* d_ws,
                              size_t ws_size, hipStream_t stream) {
  const float* hidden    = (const float*)d_in[0];
  const float* norm_w    = (const float*)d_in[1];
  const float* in_proj_w = (const float*)d_in[2];
  const float* conv_w    = (const float*)d_in[3];
  const float* conv_b    = (const float*)d_in[4];
  const float* x_proj_w  = (const float*)d_in[5];
  const float* dt_proj_w = (const float*)d_in[6];
  const float* dt_proj_b = (const float*)d_in[7];
  const float* A_log     = (const float*)d_in[8];
  const float* Dw        = (const float*)d_in[9];
  const float* out_proj_w= (const float*)d_in[10];
  float* out = (float*)d_out;

  float* ws = (float*)d_ws;
  float* hn   = ws;                                   // 4096*1024
  float* xz   = hn + (size_t)NROW * D_MODEL;          // 4096*4096
  float* xact = xz + (size_t)NROW * 2 * D_INNER;      // 4096*2048
  float* dbl  = xact + (size_t)NROW * D_INNER;        // 4096*96
  float* dtb  = dbl + (size_t)NROW * DBL_N;           // 4096*2048
  float* ys   = dtb + (size_t)NROW * D_INNER;         // 4096*2048

  // 1. RMSNorm
  rmsnorm_kernel<<<NROW, 256, 0, stream>>>(hidden, norm_w, hn);

  // 2. in_proj: xz = hn @ in_proj_w^T   (M=4096, N=4096, K=1024)
  {
    dim3 g((2 * D_INNER) / 64, NROW / 64);
    wmma_gemm_f32<0><<<g, 256, 0, stream>>>(hn, in_proj_w, xz, NROW,
                                            2 * D_INNER, D_MODEL, D_MODEL,
                                            nullptr);
  }

  // 3. conv + silu
  int nel = NROW * D_INNER;
  conv_silu_kernel<<<(nel + 255) / 256, 256, 0, stream>>>(xz, conv_w, conv_b,
                                                          xact);

  // 4. x_proj: dbl = xact @ x_proj_w^T  (M=4096, N=96, K=2048)
  {
    dim3 g((DBL_N + 63) / 64, NROW / 64);
    wmma_gemm_f32<0><<<g, 256, 0, stream>>>(xact, x_proj_w, dbl, NROW, DBL_N,
                                            D_INNER, D_INNER, nullptr);
  }

  // 5. dt = softplus(dbl[:, :64] @ dt_proj_w^T + b)  (M=4096, N=2048, K=64)
  {
    dim3 g(D_INNER / 64, NROW / 64);
    wmma_gemm_f32<1><<<g, 256, 0, stream>>>(dbl, dt_proj_w, dtb, NROW, D_INNER,
                                            DT_RANK, DBL_N, dt_proj_b);
  }

  // 6. selective scan
  scan_kernel<<<(B_SZ * D_INNER / 2) / 8, 256, 0, stream>>>(xact, dtb, dbl,
                                                            A_log, ys);

  // 7. gating (in place on ys)
  gate_kernel<<<(nel + 255) / 256, 256, 0, stream>>>(ys, xact, xz, Dw, ys);

  // 8. out = g @ out_proj_w^T + residual  (M=4096, N=1024, K=2048)
  {
    dim3 g(D_MODEL / 64, NROW / 64);
    wmma_gemm_f32<2><<<g, 256, 0, stream>>>(ys, out_proj_w, out, NROW, D_MODEL,
                                            D_INNER, D_INNER, hidden);
  }
}